// GIN_41558103556403
// MI455X (gfx1250) — compile-verified
//
#include <hip/hip_runtime.h>
#include <hip/hip_bf16.h>

typedef float v2f __attribute__((ext_vector_type(2)));
typedef float v8f __attribute__((ext_vector_type(8)));

#define DH 128
#define LDSK 132      // 16-row A tile, padded stride (words) -> conflict-free float2 reads
#define BN_ROWS 128
#define MTILES 4      // row-tiles per block: W fragments stay in registers across tiles

// ---------------------------------------------------------------- utilities
__global__ __launch_bounds__(256) void zero_f32(float* __restrict__ p, int n) {
  int i = blockIdx.x * 256 + threadIdx.x;
  int stride = gridDim.x * 256;
  for (; i < n; i += stride) p[i] = 0.0f;
}

// ------------------------------------------------- edge scatter-add (GIN aggr)
// one wave (32 lanes) per edge, float4 per lane; gather + atomics are L2-resident
__global__ __launch_bounds__(256) void gin_scatter(const float* __restrict__ x,
                                                   const int* __restrict__ src,
                                                   const int* __restrict__ dst,
                                                   float* __restrict__ aggr,
                                                   int n_edges) {
  int gid = blockIdx.x * 256 + threadIdx.x;
  int e = gid >> 5;
  if (e >= n_edges) return;
  int lane = gid & 31;
  int s = src[e], d = dst[e];
  float4 v = *((const float4*)(x + (size_t)s * DH) + lane);
  float* ap = aggr + (size_t)d * DH + (size_t)lane * 4;
  atomicAdd(ap + 0, v.x);
  atomicAdd(ap + 1, v.y);
  atomicAdd(ap + 2, v.z);
  atomicAdd(ap + 3, v.w);
}

// ------------------------------------------------- fp32 WMMA GEMM, N = K = 128
// out[M x 128] = act(in[M x 128] @ W[128 x 128] + bias)
// FUSE_AGGR: in := (1+eps)*A + aggr (GIN combine fused into the A-tile staging)
// Block: 256 threads = 8 waves; wave w owns N-tile [w*16, w*16+16).
// W fragments (128x16 per wave -> 64 VGPRs/lane) loaded ONCE, then MTILES
// row-tiles are processed reusing them: inner loop = ds_load_b64 + v_wmma only.
template <bool FUSE_AGGR, bool RELU>
__global__ __launch_bounds__(256) void gemm128_wmma(const float* __restrict__ A,
                                                    const float* __restrict__ aggr,
                                                    const float* __restrict__ epsp,
                                                    const float* __restrict__ W,
                                                    const float* __restrict__ bias,
                                                    float* __restrict__ out, int M) {
  __shared__ float As[16 * LDSK];
  const int tid = threadIdx.x;
  const int wave = tid >> 5;
  const int lane = tid & 31;
  const int l = lane & 15;   // row (A) / col (B,C) within tile
  const int hh = lane >> 4;  // half select: K pair 0/1 vs 2/3
  const int n0 = wave * 16;

  float ope = 1.0f;
  if (FUSE_AGGR) ope = 1.0f + epsp[0];

  // Preload B fragments for the whole K=128 (per-wave 16-column slice of W).
  // Lane l<16 holds rows kk,kk+1; lane l+16 holds rows kk+2,kk+3 (col n0+l).
  v2f bf[32];
#pragma unroll
  for (int ks = 0; ks < 32; ++ks) {
    int kk = ks * 4 + 2 * hh;
    bf[ks].x = W[kk * DH + n0 + l];
    bf[ks].y = W[(kk + 1) * DH + n0 + l];
  }
  const float bv = bias[n0 + l];

  for (int t = 0; t < MTILES; ++t) {
    const int m0 = (blockIdx.x * MTILES + t) * 16;
    if (m0 >= M) break;  // uniform across block: barrier-safe

    // stage 16x128 A tile to LDS (each thread: two float4s), fuse GIN combine
    for (int q = tid; q < 512; q += 256) {
      int row = q >> 5;
      int c4 = (q & 31) * 4;
      int m = m0 + row;
      float4 v = make_float4(0.f, 0.f, 0.f, 0.f);
      if (m < M) {
        v = *(const float4*)(A + (size_t)m * DH + c4);
        if (FUSE_AGGR) {
          float4 g = *(const float4*)(aggr + (size_t)m * DH + c4);
          v.x = fmaf(ope, v.x, g.x);
          v.y = fmaf(ope, v.y, g.y);
          v.z = fmaf(ope, v.z, g.z);
          v.w = fmaf(ope, v.w, g.w);
        }
      }
      *(float4*)(&As[row * LDSK + c4]) = v;
    }
    __syncthreads();

    v8f c = {};
#pragma unroll
    for (int ks = 0; ks < 32; ++ks) {
      // A 16x4 fp32 fragment: lanes 0-15 -> K=kk,kk+1 ; lanes 16-31 -> K=kk+2,kk+3
      v2f a = *(const v2f*)(&As[l * LDSK + ks * 4 + 2 * hh]);
      c = __builtin_amdgcn_wmma_f32_16x16x4_f32(false, a, false, bf[ks], (short)0,
                                                c, false, false);
    }

    // C/D layout: VGPR r -> M = r (lanes 0-15) or M = 8+r (lanes 16-31), N = lane&15
    float* orow = out + (size_t)(m0 + 8 * hh) * DH + n0 + l;
    if (m0 + 15 < M) {  // fast path: full tile, no per-row predication
#pragma unroll
      for (int r = 0; r < 8; ++r) {
        float v = c[r] + bv;
        if (RELU) v = fmaxf(v, 0.0f);
        orow[(size_t)r * DH] = v;
      }
    } else {
#pragma unroll
      for (int r = 0; r < 8; ++r) {
        int m = m0 + r + 8 * hh;
        if (m < M) {
          float v = c[r] + bv;
          if (RELU) v = fmaxf(v, 0.0f);
          out[(size_t)m * DH + n0 + l] = v;
        }
      }
    }
    __syncthreads();  // all waves done reading As before next tile restages
  }
}

// ------------------------------------------------- BatchNorm (training stats)
__global__ __launch_bounds__(128) void bn_stats(const float* __restrict__ h,
                                                float* __restrict__ sums, int M) {
  int c = threadIdx.x;  // 128 columns
  int r0 = blockIdx.x * BN_ROWS;
  int rend = min(r0 + BN_ROWS, M);
  float s = 0.f, q = 0.f;
  for (int r = r0; r < rend; ++r) {
    float v = h[(size_t)r * DH + c];
    s += v;
    q = fmaf(v, v, q);
  }
  atomicAdd(&sums[c], s);
  atomicAdd(&sums[DH + c], q);
}

__global__ __launch_bounds__(128) void bn_finalize(const float* __restrict__ sums,
                                                   const float* __restrict__ gamma,
                                                   const float* __restrict__ beta,
                                                   float* __restrict__ ss, int M) {
  int c = threadIdx.x;
  float inv = 1.0f / (float)M;
  float mu = sums[c] * inv;
  float var = fmaxf(sums[DH + c] * inv - mu * mu, 0.0f);  // biased var
  float rs = rsqrtf(var + 1e-5f);
  float sc = gamma[c] * rs;
  ss[c] = sc;
  ss[DH + c] = beta[c] - mu * sc;
}

__global__ __launch_bounds__(256) void bn_apply_relu(const float* __restrict__ h,
                                                     const float* __restrict__ ss,
                                                     float* __restrict__ xout, int n) {
  int i = blockIdx.x * 256 + threadIdx.x;
  if (i >= n) return;
  int c = i & (DH - 1);
  float v = fmaf(h[i], ss[c], ss[DH + c]);
  xout[i] = fmaxf(v, 0.0f);
}

// ------------------------------------------------- global_add_pool
__global__ __launch_bounds__(256) void pool_add(const float* __restrict__ x,
                                                const int* __restrict__ batch,
                                                float* __restrict__ g, int M) {
  int gid = blockIdx.x * 256 + threadIdx.x;
  int i = gid >> 5;
  if (i >= M) return;
  int lane = gid & 31;
  float4 v = *((const float4*)(x + (size_t)i * DH) + lane);
  float* gp = g + (size_t)batch[i] * DH + (size_t)lane * 4;
  atomicAdd(gp + 0, v.x);
  atomicAdd(gp + 1, v.y);
  atomicAdd(gp + 2, v.z);
  atomicAdd(gp + 3, v.w);
}

// ------------------------------------------------- tiny readout MLP (8 MFLOP)
__global__ __launch_bounds__(256) void readout1(const float* __restrict__ g,
                                                const float* __restrict__ W3,
                                                const float* __restrict__ b3,
                                                float* __restrict__ gr, int NG) {
  int idx = blockIdx.x * 256 + threadIdx.x;
  if (idx >= NG * DH) return;
  int i = idx >> 7, j = idx & 127;
  float acc = b3[j];
  for (int k = 0; k < DH; ++k) acc = fmaf(g[i * DH + k], W3[k * DH + j], acc);
  gr[idx] = fmaxf(acc, 0.0f);
}

__global__ __launch_bounds__(256) void readout2(const float* __restrict__ gr,
                                                const float* __restrict__ W4,
                                                const float* __restrict__ b4,
                                                float* __restrict__ out, int NG,
                                                int DOUT) {
  int idx = blockIdx.x * 256 + threadIdx.x;
  if (idx >= NG * DOUT) return;
  int i = idx / DOUT, j = idx % DOUT;
  float acc = b4[j];
  for (int k = 0; k < DH; ++k) acc = fmaf(gr[i * DH + k], W4[k * DOUT + j], acc);
  out[idx] = acc;
}

// ---------------------------------------------------------------- launcher
extern "C" void kernel_launch(void* const* d_in, const int* in_sizes, int n_in,
                              void* d_out, int out_size, void* d_ws, size_t ws_size,
                              hipStream_t stream) {
  const float* x_in   = (const float*)d_in[0];
  const int*   eidx   = (const int*)d_in[1];
  const int*   batch  = (const int*)d_in[2];
  const float* W1s    = (const float*)d_in[3];
  const float* b1s    = (const float*)d_in[4];
  const float* W2s    = (const float*)d_in[5];
  const float* b2s    = (const float*)d_in[6];
  const float* gammas = (const float*)d_in[7];
  const float* betas  = (const float*)d_in[8];
  const float* eps_g  = (const float*)d_in[9];
  const float* W3     = (const float*)d_in[10];
  const float* b3     = (const float*)d_in[11];
  const float* W4     = (const float*)d_in[12];
  const float* b4     = (const float*)d_in[13];
  float* out = (float*)d_out;
  (void)n_in; (void)ws_size;

  const int M       = in_sizes[0] / DH;  // 100000
  const int E       = in_sizes[1] / 2;   // 1600000
  const int NLAYERS = in_sizes[9];       // 3
  const int DOUT    = in_sizes[12] / DH; // 64
  const int NG      = out_size / DOUT;   // 512

  const size_t nd = (size_t)M * DH;  // node-feature buffer elements
  float* X    = (float*)d_ws;        // current features (post-layer)
  float* A1   = X + nd;              // aggr, later h2
  float* A2   = A1 + nd;             // h1
  float* g    = A2 + nd;             // pooled graph features
  float* gr   = g + (size_t)NG * DH;
  float* sums = gr + (size_t)NG * DH;  // [0..127] sum, [128..255] sumsq
  float* ss   = sums + 2 * DH;         // [0..127] scale, [128..255] shift

  const int gblocks = (M + MTILES * 16 - 1) / (MTILES * 16);

  for (int l = 0; l < NLAYERS; ++l) {
    const float* xc = (l == 0) ? x_in : X;
    zero_f32<<<2048, 256, 0, stream>>>(A1, (int)nd);
    gin_scatter<<<(E * 32 + 255) / 256, 256, 0, stream>>>(xc, eidx, eidx + E, A1, E);
    // h1 = relu(((1+eps)x + aggr) @ W1 + b1)
    gemm128_wmma<true, true><<<gblocks, 256, 0, stream>>>(
        xc, A1, eps_g + l, W1s + (size_t)l * DH * DH, b1s + (size_t)l * DH, A2, M);
    // h2 = h1 @ W2 + b2
    gemm128_wmma<false, false><<<gblocks, 256, 0, stream>>>(
        A2, nullptr, nullptr, W2s + (size_t)l * DH * DH, b2s + (size_t)l * DH, A1, M);
    zero_f32<<<1, 256, 0, stream>>>(sums, 2 * DH);
    bn_stats<<<(M + BN_ROWS - 1) / BN_ROWS, 128, 0, stream>>>(A1, sums, M);
    bn_finalize<<<1, 128, 0, stream>>>(sums, gammas + (size_t)l * DH,
                                       betas + (size_t)l * DH, ss, M);
    bn_apply_relu<<<((int)nd + 255) / 256, 256, 0, stream>>>(A1, ss, X, (int)nd);
  }

  zero_f32<<<64, 256, 0, stream>>>(g, NG * DH);
  pool_add<<<(M * 32 + 255) / 256, 256, 0, stream>>>(X, batch, g, M);
  readout1<<<(NG * DH + 255) / 256, 256, 0, stream>>>(g, W3, b3, gr, NG);
  readout2<<<(NG * DOUT + 255) / 256, 256, 0, stream>>>(gr, W4, b4, out, NG, DOUT);
}